// RandlaNet_28355374088201
// MI455X (gfx1250) — compile-verified
//
#include <hip/hip_runtime.h>
#include <hip/hip_bf16.h>

typedef __attribute__((ext_vector_type(2))) float v2f;
typedef __attribute__((ext_vector_type(8))) float v8f;

__device__ __forceinline__ int imin(int a, int b) { return a < b ? a : b; }

// ---------------------------------------------------------------------------
// WMMA f32 GEMM:  Y[M x Nc] = X[M x Kd] @ W[Kd x Nc] (+ bias), row-major.
// One wave computes one 16x16 tile via V_WMMA_F32_16X16X4_F32.
// 4 waves per block tile along M. ldY allows writing into a column slice of a
// wider buffer (used to fuse the enc->feat concat).
// Out-of-range rows/cols/k are handled by clamped addressing + multiplicative
// masks so the inner loop has NO divergent branches (loads stay unconditional;
// EXEC stays all-ones for every WMMA).
// ---------------------------------------------------------------------------
__global__ __launch_bounds__(128)
void gemm_wmma_f32(const float* __restrict__ X, int ldX,
                   const float* __restrict__ W,
                   const float* __restrict__ bias,
                   float* __restrict__ Y, int ldY,
                   int M, int Kd, int Nc) {
  const int wave = threadIdx.x >> 5;
  const int lane = threadIdx.x & 31;
  const int half = lane >> 4;      // 0: lanes 0-15, 1: lanes 16-31
  const int l    = lane & 15;
  const int tileM = (blockIdx.x * 4 + wave) * 16;
  const int tileN = blockIdx.y * 16;
  if (tileM >= M) return;          // wave-uniform exit

  const int   rowA  = imin(tileM + l, M - 1);
  const float rmask = (tileM + l < M) ? 1.f : 0.f;
  const int   colB  = imin(tileN + l, Nc - 1);
  const float cmask = (tileN + l < Nc) ? 1.f : 0.f;

  const float* xr = X + (size_t)rowA * ldX + half * 2;   // A: K = half*2 + {0,1}
  const float* wc = W + (size_t)(half * 2) * Nc + colB;  // B: rows K = half*2 + {0,1}
  const size_t wstep = (size_t)Nc;

  v8f acc = {};
  int k = 0;
  // Main loop: 16 K-values (4 WMMAs) per iteration, branch-free.
  for (; k + 16 <= Kd; k += 16) {
#pragma unroll
    for (int u = 0; u < 4; ++u) {
      const int kk = k + u * 4;
      v2f a, b;
      a.x = xr[kk]     * rmask;
      a.y = xr[kk + 1] * rmask;
      b.x = wc[(size_t)kk * wstep]       * cmask;
      b.y = wc[(size_t)(kk + 1) * wstep] * cmask;
      acc = __builtin_amdgcn_wmma_f32_16x16x4_f32(false, a, false, b,
                                                  (short)0, acc, false, false);
    }
  }
  // Tail (Kd % 16): clamped addresses + value masks, still branch-free.
  for (; k < Kd; k += 4) {
    const int kk  = k + half * 2;
    const int k0  = imin(kk,     Kd - 1);
    const int k1  = imin(kk + 1, Kd - 1);
    const float m0 = (kk     < Kd) ? rmask : 0.f;
    const float m1 = (kk + 1 < Kd) ? rmask : 0.f;
    const float n0 = (kk     < Kd) ? cmask : 0.f;
    const float n1 = (kk + 1 < Kd) ? cmask : 0.f;
    v2f a, b;
    a.x = X[(size_t)rowA * ldX + k0] * m0;
    a.y = X[(size_t)rowA * ldX + k1] * m1;
    b.x = W[(size_t)k0 * wstep + colB] * n0;
    b.y = W[(size_t)k1 * wstep + colB] * n1;
    acc = __builtin_amdgcn_wmma_f32_16x16x4_f32(false, a, false, b,
                                                (short)0, acc, false, false);
  }

  if (tileN + l >= Nc) return;
  const float bv = bias ? bias[colB] : 0.f;
#pragma unroll
  for (int v = 0; v < 8; ++v) {    // C layout: VGPR v -> row M = v + 8*half
    const int r = tileM + half * 8 + v;
    if (r < M) Y[(size_t)r * ldY + colB] = acc[v] + bv;
  }
}

// ---------------------------------------------------------------------------
// Per-column mean / biased variance over M rows (one block per column).
// ---------------------------------------------------------------------------
__global__ void col_stats(const float* __restrict__ Y, int ldY, int M,
                          float* __restrict__ mean, float* __restrict__ var_) {
  const int c = blockIdx.x;
  float s = 0.f, s2 = 0.f;
  for (int r = threadIdx.x; r < M; r += blockDim.x) {
    const float v = Y[(size_t)r * ldY + c];
    s += v; s2 += v * v;
  }
  __shared__ float sh[256], sh2[256];
  const int t = threadIdx.x;
  sh[t] = s; sh2[t] = s2;
  __syncthreads();
  for (int st = 128; st > 0; st >>= 1) {
    if (t < st) { sh[t] += sh[t + st]; sh2[t] += sh2[t + st]; }
    __syncthreads();
  }
  if (t == 0) {
    const float mu = sh[0] / (float)M;
    mean[c] = mu;
    var_[c] = sh2[0] / (float)M - mu * mu;
  }
}

// ---------------------------------------------------------------------------
// BatchNorm-style normalize (+ optional leaky relu 0.2), in place.
// ---------------------------------------------------------------------------
__global__ void bn_act(float* __restrict__ Y, int ldY, int M, int Nc,
                       const float* __restrict__ mean, const float* __restrict__ var_,
                       const float* __restrict__ g, const float* __restrict__ bt,
                       int act) {
  const long t = (long)blockIdx.x * blockDim.x + threadIdx.x;
  if (t >= (long)M * Nc) return;
  const int  c = (int)(t % Nc);
  const long r = t / Nc;
  float v = Y[(size_t)r * ldY + c];
  v = (v - mean[c]) * rsqrtf(var_[c] + 1e-5f) * g[c] + bt[c];
  if (act) v = v > 0.f ? v : 0.2f * v;
  Y[(size_t)r * ldY + c] = v;
}

// ---------------------------------------------------------------------------
// Brute-force 16-NN (self set == ref set). LDS-tiled refs; per-thread top-16
// kept unsorted with replace-max (neighbor order is irrelevant downstream:
// everything reduces over the K axis).
// ---------------------------------------------------------------------------
#define KNN_TILE 256
__global__ void knn16_kernel(const float* __restrict__ pos, int n,
                             int* __restrict__ out) {
  const int q = blockIdx.x * blockDim.x + threadIdx.x;
  float qx = 0.f, qy = 0.f, qz = 0.f;
  if (q < n) { qx = pos[q * 3]; qy = pos[q * 3 + 1]; qz = pos[q * 3 + 2]; }
  float bd[16]; int bi[16];
#pragma unroll
  for (int s = 0; s < 16; ++s) { bd[s] = 3.4e38f; bi[s] = (q < n) ? q : 0; }
  float worst = 3.4e38f; int wslot = 0;
  __shared__ float sx[KNN_TILE], sy[KNN_TILE], sz[KNN_TILE];
  for (int base = 0; base < n; base += KNN_TILE) {
    const int lim = (KNN_TILE < n - base) ? KNN_TILE : (n - base);
    __syncthreads();
    if ((int)threadIdx.x < lim) {
      const int j = base + threadIdx.x;
      sx[threadIdx.x] = pos[j * 3];
      sy[threadIdx.x] = pos[j * 3 + 1];
      sz[threadIdx.x] = pos[j * 3 + 2];
    }
    __syncthreads();
    if (q < n) {
      for (int t = 0; t < lim; ++t) {
        const float dx = qx - sx[t], dy = qy - sy[t], dz = qz - sz[t];
        const float d = dx * dx + dy * dy + dz * dz;
        if (d < worst) {
          bd[wslot] = d; bi[wslot] = base + t;
          worst = bd[0]; wslot = 0;
#pragma unroll
          for (int s = 1; s < 16; ++s)
            if (bd[s] > worst) { worst = bd[s]; wslot = s; }
        }
      }
    }
  }
  if (q < n) {
#pragma unroll
    for (int s = 0; s < 16; ++s) out[q * 16 + s] = bi[s];
  }
}

// 1-NN: queries qpos[nq], refs rpos[nr].
__global__ void nn1_kernel(const float* __restrict__ qpos, int nq,
                           const float* __restrict__ rpos, int nr,
                           int* __restrict__ out) {
  const int q = blockIdx.x * blockDim.x + threadIdx.x;
  float qx = 0.f, qy = 0.f, qz = 0.f;
  if (q < nq) { qx = qpos[q * 3]; qy = qpos[q * 3 + 1]; qz = qpos[q * 3 + 2]; }
  float best = 3.4e38f; int bidx = 0;
  __shared__ float sx[KNN_TILE], sy[KNN_TILE], sz[KNN_TILE];
  for (int base = 0; base < nr; base += KNN_TILE) {
    const int lim = (KNN_TILE < nr - base) ? KNN_TILE : (nr - base);
    __syncthreads();
    if ((int)threadIdx.x < lim) {
      const int j = base + threadIdx.x;
      sx[threadIdx.x] = rpos[j * 3];
      sy[threadIdx.x] = rpos[j * 3 + 1];
      sz[threadIdx.x] = rpos[j * 3 + 2];
    }
    __syncthreads();
    if (q < nq) {
      for (int t = 0; t < lim; ++t) {
        const float dx = qx - sx[t], dy = qy - sy[t], dz = qz - sz[t];
        const float d = dx * dx + dy * dy + dz * dz;
        if (d < best) { best = d; bidx = base + t; }
      }
    }
  }
  if (q < nq) out[q] = bidx;
}

// rel = [pos_i(3), pos_j(3), d(3), |d|(1)] per (i, k). One thread per (i,k).
__global__ void build_rel(const float* __restrict__ pos, const int* __restrict__ nbr,
                          float* __restrict__ rel, int n) {
  const long t = (long)blockIdx.x * blockDim.x + threadIdx.x;
  if (t >= (long)n * 16) return;
  const int i = (int)(t >> 4);
  const int j = nbr[t];
  const float ax = pos[i * 3], ay = pos[i * 3 + 1], az = pos[i * 3 + 2];
  const float bx = pos[j * 3], by = pos[j * 3 + 1], bz = pos[j * 3 + 2];
  const float dx = bx - ax, dy = by - ay, dz = bz - az;
  const float dist = sqrtf(dx * dx + dy * dy + dz * dz);
  float* o = rel + (size_t)t * 10;
  o[0] = ax; o[1] = ay; o[2] = az;
  o[3] = bx; o[4] = by; o[5] = bz;
  o[6] = dx; o[7] = dy; o[8] = dz; o[9] = dist;
}

// feat[:, 0:ch] = x[nbr] with row stride ldF (concat with enc half).
__global__ void gather_nbr_feat(const float* __restrict__ x, const int* __restrict__ nbr,
                                float* __restrict__ feat, int n, int ch, int ldF) {
  const long t = (long)blockIdx.x * blockDim.x + threadIdx.x;
  if (t >= (long)n * 16 * ch) return;
  const int  c  = (int)(t % ch);
  const long ik = t / ch;
  const int  j  = nbr[ik];
  feat[ik * (size_t)ldF + c] = x[(size_t)j * ch + c];
}

// out[i,c] = sum_k softmax_k(logit[i,k,c]) * feat[i,k,c]. One thread per (i,c).
__global__ void att_softmax_sum(const float* __restrict__ feat,
                                const float* __restrict__ logit,
                                float* __restrict__ out, int n, int c) {
  const long t = (long)blockIdx.x * blockDim.x + threadIdx.x;
  if (t >= (long)n * c) return;
  const int  ch = (int)(t % c);
  const long i  = t / c;
  const float* L = logit + (size_t)(i * 16) * c + ch;
  const float* F = feat + (size_t)(i * 16) * c + ch;
  float mx = -3.4e38f;
#pragma unroll
  for (int k = 0; k < 16; ++k) mx = fmaxf(mx, L[(size_t)k * c]);
  float se = 0.f, acc = 0.f;
#pragma unroll
  for (int k = 0; k < 16; ++k) {
    const float e = __expf(L[(size_t)k * c] - mx);
    se += e;
    acc += e * F[(size_t)k * c];
  }
  out[(size_t)i * c + ch] = acc / se;
}

__global__ void add_lrelu(const float* __restrict__ a, const float* __restrict__ b,
                          float* __restrict__ o, long n) {
  const long t = (long)blockIdx.x * blockDim.x + threadIdx.x;
  if (t >= n) return;
  const float v = a[t] + b[t];
  o[t] = v > 0.f ? v : 0.2f * v;
}

// out[i] = concat(xc[idx[i]] (cc ch), xs[i] (cs ch)).
__global__ void gather_concat(const float* __restrict__ xc, const int* __restrict__ idx,
                              const float* __restrict__ xs, float* __restrict__ out,
                              int n, int cc, int cs) {
  const int ct = cc + cs;
  const long t = (long)blockIdx.x * blockDim.x + threadIdx.x;
  if (t >= (long)n * ct) return;
  const int  ch = (int)(t % ct);
  const long i  = t / ct;
  out[t] = (ch < cc) ? xc[(size_t)idx[i] * cc + ch]
                     : xs[i * (size_t)cs + (ch - cc)];
}

__global__ void log_softmax_rows(const float* __restrict__ X, float* __restrict__ out,
                                 int n, int C) {
  const int i = blockIdx.x * blockDim.x + threadIdx.x;
  if (i >= n) return;
  const float* r = X + (size_t)i * C;
  float mx = -3.4e38f;
  for (int c = 0; c < C; ++c) mx = fmaxf(mx, r[c]);
  float se = 0.f;
  for (int c = 0; c < C; ++c) se += __expf(r[c] - mx);
  const float lse = mx + logf(se);
  for (int c = 0; c < C; ++c) out[(size_t)i * C + c] = r[c] - lse;
}

__global__ void write_seed(float* __restrict__ o, int n) {
  const int i = blockIdx.x * blockDim.x + threadIdx.x;
  if (i < n) o[i] = (float)i;
}

// ---------------------------------------------------------------------------
// Host-side orchestration
// ---------------------------------------------------------------------------
namespace {

struct SM  { const float *W, *b, *g, *bt; };
struct LFA { SM enc; const float* attW; SM post; };
struct BLK { SM shortcut, mlp1; LFA lfa1, lfa2; SM mlp2; };

struct Stack {
  char* base; size_t off, cap;
  void* raw(size_t bytes) {
    size_t a = (off + 255) & ~(size_t)255;
    off = a + bytes;
    return base + a;
  }
  float* f(size_t n) { return (float*)raw(n * sizeof(float)); }
  int*   i(size_t n) { return (int*)raw(n * sizeof(int)); }
};

inline unsigned gdiv(long n) { return (unsigned)((n + 255) / 256); }

inline void launch_gemm(hipStream_t s, const float* X, int ldX, const float* W,
                        const float* bias, float* Y, int ldY, int M, int Kd, int Nc) {
  dim3 g((unsigned)((M + 63) / 64), (unsigned)((Nc + 15) / 16));
  gemm_wmma_f32<<<g, 128, 0, s>>>(X, ldX, W, bias, Y, ldY, M, Kd, Nc);
}

inline void apply_sm(hipStream_t s, const SM& p, const float* X, int M, int din,
                     int dout, float* Y, int ldY, float* mean, float* var_, int act) {
  launch_gemm(s, X, din, p.W, p.b, Y, ldY, M, din, dout);
  col_stats<<<dout, 256, 0, s>>>(Y, ldY, M, mean, var_);
  const long tot = (long)M * dout;
  bn_act<<<gdiv(tot), 256, 0, s>>>(Y, ldY, M, dout, mean, var_, p.g, p.bt, act);
}

inline void apply_lfa(hipStream_t s, const LFA& p, const float* xin, const float* pos,
                      const int* nbr, int n, int c, float* outY, Stack& ws,
                      float* mean, float* var_) {
  const size_t mark = ws.off;
  const long M = (long)n * 16;
  float* rel = ws.f((size_t)M * 10);
  build_rel<<<gdiv(M), 256, 0, s>>>(pos, nbr, rel, n);
  float* feat = ws.f((size_t)M * c);
  // enc -> feat columns [c/2, c)
  apply_sm(s, p.enc, rel, (int)M, 10, c / 2, feat + c / 2, c, mean, var_, 1);
  // x[nbr] -> feat columns [0, c/2)
  gather_nbr_feat<<<gdiv(M * (c / 2)), 256, 0, s>>>(xin, nbr, feat, n, c / 2, c);
  float* att = ws.f((size_t)M * c);
  launch_gemm(s, feat, c, p.attW, nullptr, att, c, (int)M, c, c);
  float* pooled = ws.f((size_t)n * c);
  att_softmax_sum<<<gdiv((long)n * c), 256, 0, s>>>(feat, att, pooled, n, c);
  apply_sm(s, p.post, pooled, n, c, c, outY, c, mean, var_, 1);
  ws.off = mark;
}

inline void apply_block(hipStream_t s, const BLK& b, const float* x, const float* pos,
                        int n, int din, int dout, float* outY, Stack& ws,
                        float* mean, float* var_) {
  const size_t mark = ws.off;
  int* nbr = ws.i((size_t)n * 16);
  knn16_kernel<<<gdiv(n), 256, 0, s>>>(pos, n, nbr);
  float* sc = ws.f((size_t)n * dout);
  apply_sm(s, b.shortcut, x, n, din, dout, sc, dout, mean, var_, 0);
  float* hA = ws.f((size_t)n * (dout / 8));
  apply_sm(s, b.mlp1, x, n, din, dout / 8, hA, dout / 8, mean, var_, 1);
  const int c1 = dout / 4, c2 = dout / 2;
  float* hB = ws.f((size_t)n * c1);
  apply_lfa(s, b.lfa1, hA, pos, nbr, n, c1, hB, ws, mean, var_);
  float* hC = ws.f((size_t)n * c2);
  apply_lfa(s, b.lfa2, hB, pos, nbr, n, c2, hC, ws, mean, var_);
  float* hm = ws.f((size_t)n * dout);
  apply_sm(s, b.mlp2, hC, n, c2, dout, hm, dout, mean, var_, 0);
  const long tot = (long)n * dout;
  add_lrelu<<<gdiv(tot), 256, 0, s>>>(hm, sc, outY, tot);
  ws.off = mark;
}

} // namespace

extern "C" void kernel_launch(void* const* d_in, const int* in_sizes, int n_in,
                              void* d_out, int out_size, void* d_ws, size_t ws_size,
                              hipStream_t stream) {
  (void)in_sizes; (void)n_in; (void)out_size;
  const int N = 65536, N1 = 16384, N2 = 4096, N3 = 1024, N4 = 256;

  const float* x   = (const float*)d_in[0];
  const float* pos = (const float*)d_in[1];

  // Consume params in Python-dict insertion order.
  int pi = 2;
  auto nf = [&]() -> const float* { return (const float*)d_in[pi++]; };
  auto loadSM  = [&](SM& m)  { m.W = nf(); m.b = nf(); m.g = nf(); m.bt = nf(); };
  auto loadLFA = [&](LFA& l) { loadSM(l.enc); l.attW = nf(); loadSM(l.post); };
  auto loadBLK = [&](BLK& b) {
    loadSM(b.shortcut); loadSM(b.mlp1);
    loadLFA(b.lfa1); loadLFA(b.lfa2);
    loadSM(b.mlp2);
  };

  const float* fc0W = nf(); const float* fc0b = nf();
  BLK b1, b2, b3, b4;
  loadBLK(b1); loadBLK(b2); loadBLK(b3); loadBLK(b4);
  SM summit; loadSM(summit);
  SM sem_fp[4], inst_fp[4];
  for (int i = 0; i < 4; ++i) loadSM(sem_fp[i]);   // fp4, fp3, fp2, fp1
  for (int i = 0; i < 4; ++i) loadSM(inst_fp[i]);
  SM classif1, classif2; loadSM(classif1); loadSM(classif2);
  const float* fcoW = nf(); const float* fcob = nf();

  Stack ws{(char*)d_ws, 0, ws_size};

  // Persistent buffers
  float* mean = ws.f(1024);
  float* var_ = ws.f(1024);
  float* h1 = ws.f((size_t)N * 32);
  float* h2 = ws.f((size_t)N1 * 128);
  float* h3 = ws.f((size_t)N2 * 256);
  float* m_ = ws.f((size_t)N4 * 512);
  const size_t mark0 = ws.off;

  // Encoder
  {
    float* h0 = ws.f((size_t)N * 32);
    launch_gemm(stream, x, 6, fc0W, fc0b, h0, 32, N, 6, 32);
    apply_block(stream, b1, h0, pos, N, 32, 32, h1, ws, mean, var_);
  }
  ws.off = mark0;
  apply_block(stream, b2, h1, pos, N1, 32, 128, h2, ws, mean, var_);
  ws.off = mark0;
  apply_block(stream, b3, h2, pos, N2, 128, 256, h3, ws, mean, var_);
  ws.off = mark0;
  {
    float* h4 = ws.f((size_t)N3 * 512);
    apply_block(stream, b4, h3, pos, N3, 256, 512, h4, ws, mean, var_);
    apply_sm(stream, summit, h4, N4, 512, 512, m_, 512, mean, var_, 1); // h4d = h4[:256]
  }
  ws.off = mark0;

  // Shared 1-NN upsampling indices (identical for sem & inst decoders)
  int* idx4 = ws.i(N3);
  int* idx3 = ws.i(N2);
  int* idx2 = ws.i(N1);
  int* idx1 = ws.i(N);
  nn1_kernel<<<gdiv(N3), 256, 0, stream>>>(pos, N3, pos, N4, idx4);
  nn1_kernel<<<gdiv(N2), 256, 0, stream>>>(pos, N2, pos, N3, idx3);
  nn1_kernel<<<gdiv(N1), 256, 0, stream>>>(pos, N1, pos, N2, idx2);
  nn1_kernel<<<gdiv(N),  256, 0, stream>>>(pos, N,  pos, N1, idx1);

  auto run_decoder = [&](const SM* fp, float* finalOut /* N x 32 */) {
    const size_t mk = ws.off;
    float* cat4 = ws.f((size_t)N3 * 768);
    gather_concat<<<gdiv((long)N3 * 768), 256, 0, stream>>>(m_, idx4, h3, cat4, N3, 512, 256);
    float* y4 = ws.f((size_t)N3 * 256);
    apply_sm(stream, fp[0], cat4, N3, 768, 256, y4, 256, mean, var_, 1);

    float* cat3 = ws.f((size_t)N2 * 384);
    gather_concat<<<gdiv((long)N2 * 384), 256, 0, stream>>>(y4, idx3, h2, cat3, N2, 256, 128);
    float* y3 = ws.f((size_t)N2 * 128);
    apply_sm(stream, fp[1], cat3, N2, 384, 128, y3, 128, mean, var_, 1);

    float* cat2 = ws.f((size_t)N1 * 160);
    gather_concat<<<gdiv((long)N1 * 160), 256, 0, stream>>>(y3, idx2, h1, cat2, N1, 128, 32);
    float* y2 = ws.f((size_t)N1 * 32);
    apply_sm(stream, fp[2], cat2, N1, 160, 32, y2, 32, mean, var_, 1);

    float* cat1 = ws.f((size_t)N * 64);
    gather_concat<<<gdiv((long)N * 64), 256, 0, stream>>>(y2, idx1, h1, cat1, N, 32, 32);
    apply_sm(stream, fp[3], cat1, N, 64, 32, finalOut, 32, mean, var_, 1);
    ws.off = mk;
  };

  float* out_f = (float*)d_out;

  // Semantic branch -> log_softmax logits into d_out[0 : N*13)
  {
    const size_t mk = ws.off;
    float* semf = ws.f((size_t)N * 32);
    run_decoder(sem_fp, semf);
    float* c1o = ws.f((size_t)N * 64);
    apply_sm(stream, classif1, semf, N, 32, 64, c1o, 64, mean, var_, 1);
    float* c2o = ws.f((size_t)N * 32);
    apply_sm(stream, classif2, c1o, N, 64, 32, c2o, 32, mean, var_, 1);
    float* logits = ws.f((size_t)N * 13);
    launch_gemm(stream, c2o, 32, fcoW, fcob, logits, 13, N, 32, 13);
    log_softmax_rows<<<gdiv(N), 256, 0, stream>>>(logits, out_f, N, 13);
    ws.off = mk;
  }

  // Instance branch -> d_out[N*13 : N*13 + N*32)
  run_decoder(inst_fp, out_f + (size_t)N * 13);

  // seed_idx = arange(256) -> tail of d_out
  write_seed<<<1, 256, 0, stream>>>(out_f + (size_t)N * 13 + (size_t)N * 32, N4);
}